// EAG_59236188946465
// MI455X (gfx1250) — compile-verified
//
#include <hip/hip_runtime.h>
#include <stdint.h>

// Problem constants (from the reference): B=2048, N=1024, C=128, M=64
#define B_ 2048
#define N_ 1024
#define C_ 128
#define M_ 64

typedef __attribute__((ext_vector_type(2))) float v2f;
typedef __attribute__((ext_vector_type(4))) float v4f;
typedef __attribute__((ext_vector_type(8))) float v8f;

// ---------------------------------------------------------------------------
// Kernel 1: streaming copy with erase. One block handles (batch b, 64 rows).
// ~2 GB of the 2.07 GB total traffic lives here -> NT 128-bit streaming.
// Erased rows are skipped entirely (kernel 2 owns those rows' writes).
// ---------------------------------------------------------------------------
#define TILE_N 64

__global__ __launch_bounds__(256) void eag_copy_erase(
    const float* __restrict__ src, const int* __restrict__ indices,
    float* __restrict__ out)
{
    __shared__ int s_idx[M_];
    __shared__ int s_erased[TILE_N];

    const int b  = blockIdx.x;
    const int n0 = blockIdx.y * TILE_N;
    const int t  = threadIdx.x;

    if (t < M_) s_idx[t] = indices[b * M_ + t];
    __syncthreads();

    if (t < TILE_N) {
        const int n = n0 + t;
        int er = 0;
        #pragma unroll 8
        for (int m = 0; m < M_; ++m) er |= (s_idx[m] == n);
        s_erased[t] = er;
    }
    __syncthreads();

    const size_t base = ((size_t)b * N_ + n0) * C_;
    const v4f* __restrict__ src4 = (const v4f*)(src + base);
    v4f* __restrict__ out4       = (v4f*)(out + base);

    // TILE_N * C_ / 4 = 2048 float4 per block; 256 threads -> 8 each.
    #pragma unroll
    for (int k = 0; k < 8; ++k) {
        const int v   = t + (k << 8);
        const int row = v >> 5;              // 32 float4 per row (C=128)
        if (k < 7)
            __builtin_prefetch(&src4[t + ((k + 1) << 8)], 0, 0); // global_prefetch_b8
        if (!s_erased[row]) {
            v4f d = __builtin_nontemporal_load(&src4[v]);   // TH=NT load
            __builtin_nontemporal_store(d, &out4[v]);       // TH=NT store
        }
    }
}

// ---------------------------------------------------------------------------
// Kernel 2: scatter-add of tar into erased rows via WMMA.
//   D(64x128) = S(64x64) @ tar(64x128),  S[i][j] = (idx[i]==idx[j])
// Row i of D is the full duplicate-accumulated value for memory row idx[i];
// duplicate i's store identical bits, so stores need no leader election and
// EXEC stays all-ones across the WMMAs. One 128-thread block (4 waves) per
// batch; wave w owns output rows [16w, 16w+16), loops over 8 column tiles.
// tar[b] (32 KB) is staged to padded LDS with async global->LDS B128 copies.
// ---------------------------------------------------------------------------
#define LDSTRIDE 132   // floats per LDS row (128 + 4 pad, keeps 16B alignment)

__global__ __launch_bounds__(128) void eag_scatter_wmma(
    const float* __restrict__ tar, const int* __restrict__ indices,
    float* __restrict__ out)
{
    __shared__ int   s_idx[M_];
    __shared__ float s_tar[M_ * LDSTRIDE];

    const int b    = blockIdx.x;
    const int t    = threadIdx.x;   // 0..127
    const int lane = t & 31;
    const int wave = t >> 5;        // 0..3
    const int hi   = lane >> 4;     // lane half: 0 -> lanes 0-15, 1 -> 16-31
    const int l    = lane & 15;

    if (t < M_) s_idx[t] = indices[b * M_ + t];

    // ---- async stage tar[b] (64x128 f32) into LDS (ASYNCcnt path) ----
    const float* gtar = tar + (size_t)b * (M_ * C_);
    const uint32_t lds_base = (uint32_t)(size_t)(&s_tar[0]); // AS3 byte offset
    #pragma unroll
    for (int k = 0; k < 16; ++k) {
        const int v   = t + (k << 7);        // float4 index 0..2047
        const int row = v >> 5;
        const int c4  = v & 31;
        const uint32_t lds_off = lds_base + (uint32_t)(row * (LDSTRIDE * 4) + c4 * 16);
        const unsigned long long gaddr =
            (unsigned long long)(size_t)(gtar + (size_t)v * 4);
        asm volatile("global_load_async_to_lds_b128 %0, %1, off"
                     :: "v"(lds_off), "v"(gaddr) : "memory");
    }
    asm volatile("s_wait_asynccnt 0" ::: "memory");
    __syncthreads();

    // A-matrix (16x4 f32) layout: lanes 0-15 hold M=0..15 {VGPR0=K0,VGPR1=K1},
    // lanes 16-31 hold M=0..15 {VGPR0=K2,VGPR1=K3}  (ISA 7.12.2).
    const int i_row  = wave * 16 + l;        // this lane's A row
    const int my_idx = s_idx[i_row];

    for (int tn = 0; tn < 8; ++tn) {
        v8f acc = (v8f)0.0f;
        const int n = tn * 16 + l;           // B/D column for this lane

        #pragma unroll
        for (int k0 = 0; k0 < M_; k0 += 4) {
            const int ka = k0 + (hi ? 2 : 0);
            v2f a, bb;
            a.x  = (s_idx[ka]     == my_idx) ? 1.0f : 0.0f;   // S[i][ka]
            a.y  = (s_idx[ka + 1] == my_idx) ? 1.0f : 0.0f;   // S[i][ka+1]
            bb.x = s_tar[ka * LDSTRIDE + n];                   // tar[ka][n]
            bb.y = s_tar[(ka + 1) * LDSTRIDE + n];             // tar[ka+1][n]
#if __has_builtin(__builtin_amdgcn_wmma_f32_16x16x4_f32)
            acc = __builtin_amdgcn_wmma_f32_16x16x4_f32(
                false, a, false, bb, (short)0, acc, false, false);
#else
            // Fallback (not expected on gfx1250): keeps TU compiling.
            acc[0] += a.x * bb.x + a.y * bb.y;
#endif
        }

        // D layout: VGPR v -> row 16*wave + v + (hi?8:0), col n (ISA 7.12.2).
        #pragma unroll
        for (int v = 0; v < 8; ++v) {
            const int i = wave * 16 + v + (hi ? 8 : 0);
            const int r = s_idx[i];          // destination memory row
            out[((size_t)b * N_ + r) * C_ + n] = acc[v];
        }
    }
}

// ---------------------------------------------------------------------------
extern "C" void kernel_launch(void* const* d_in, const int* in_sizes, int n_in,
                              void* d_out, int out_size, void* d_ws, size_t ws_size,
                              hipStream_t stream) {
    (void)in_sizes; (void)n_in; (void)d_ws; (void)ws_size; (void)out_size;
    const float* src = (const float*)d_in[0];   // (B,N,C) f32
    const float* tar = (const float*)d_in[1];   // (B,M,C) f32
    const int*   idx = (const int*)d_in[2];     // (B,M)   i32
    float* out = (float*)d_out;                 // (B,N,C) f32

    dim3 g1(B_, N_ / TILE_N);
    eag_copy_erase<<<g1, 256, 0, stream>>>(src, idx, out);      // non-erased rows
    eag_scatter_wmma<<<B_, 128, 0, stream>>>(tar, idx, out);    // erased rows
}